// BEVRasterizationBlock_14937896255876
// MI455X (gfx1250) — compile-verified
//
#include <hip/hip_runtime.h>
#include <cstdint>

#define HH 300
#define WW 400
#define HWSZ (HH * WW)

__device__ __forceinline__ unsigned encf(float f) {
    unsigned u = __float_as_uint(f);
    return (u & 0x80000000u) ? ~u : (u | 0x80000000u);
}
__device__ __forceinline__ float decf(unsigned e) {
    unsigned u = (e & 0x80000000u) ? (e & 0x7fffffffu) : ~e;
    return __uint_as_float(u);
}
__device__ __forceinline__ float clamp01(float v) {
    return fminf(fmaxf(v, 0.0f), 1.0f);
}

__global__ void zero_kernel(float* __restrict__ out, int n) {
    int i = blockIdx.x * blockDim.x + threadIdx.x;
    if (i < n) out[i] = 0.0f;
}

// Lidar scatter: stream points through LDS via the CDNA5 async-copy path,
// then scatter with L2-resident global atomics.
__global__ void lidar_scatter_kernel(const float4* __restrict__ pts,
                                     float* __restrict__ out, int n) {
    __shared__ float4 stage[256];
    int idx = blockIdx.x * blockDim.x + threadIdx.x;
    if (idx < n) {
        unsigned lds_addr = (unsigned)(unsigned long long)(void*)&stage[threadIdx.x];
        unsigned long long gaddr = (unsigned long long)(const void*)(pts + idx);
        // gfx1250 async global->LDS copy (ASYNCcnt-tracked, per-lane LDS address)
        asm volatile("global_load_async_to_lds_b128 %0, %1, off"
                     :: "v"(lds_addr), "v"(gaddr) : "memory");
        asm volatile("s_wait_asynccnt 0" ::: "memory");
        float4 p = stage[threadIdx.x];

        float x = p.x, y = p.y, z = p.z, inten = p.w;
        if (x >= -20.0f && x < 20.0f && y >= -10.0f && y < 30.0f) {
            int px = (int)((x - (-20.0f)) / 0.1f);   // trunc, matches astype(int32)
            int py = (int)((y - (-10.0f)) / 0.1f);
            px = min(max(px, 0), WW - 1);
            py = min(max(py, 0), HH - 1);
            int cell = py * WW + px;
            atomicMax((unsigned*)(out + cell), encf(z));      // ch0: encoded max z
            atomicAdd(out + HWSZ + cell, inten);              // ch1: intensity sum
            atomicAdd(out + 2 * HWSZ + cell, 1.0f);           // ch2: count
        }
    }
}

__global__ void finalize_kernel(float* __restrict__ out) {
    int i = blockIdx.x * blockDim.x + threadIdx.x;
    if (i >= HWSZ) return;
    unsigned e = __float_as_uint(out[i]);
    float hmax = (e == 0u) ? 0.0f : decf(e);       // empty cell: -inf -> 0.0
    float h = clamp01((hmax - (-3.0f)) / (4.0f - (-3.0f)));
    float cnt = out[2 * HWSZ + i];
    float isum = out[HWSZ + i];
    float imean = (cnt > 0.0f) ? (isum / fmaxf(cnt, 1.0f)) : 0.0f;
    float ich = clamp01(imean / 255.0f);
    float dch = clamp01(log1pf(cnt) / logf(1.0f + 128.0f));
    out[i] = h;
    out[HWSZ + i] = ich;
    out[2 * HWSZ + i] = dch;
}

// One block per segment, thread k = sample index along the DDA.
__global__ void polyline_kernel(const float2* __restrict__ pts, int npts,
                                float* __restrict__ out_ch,
                                const float* __restrict__ ego, int do_xform) {
    int s = blockIdx.x;
    int k = threadIdx.x;
    if (s >= npts - 1) return;

    float2 p0 = pts[s];
    float2 p1 = pts[s + 1];
    if (do_xform) {
        float e0 = ego[0], e1 = ego[1], yaw = ego[2];
        float cy = cosf(-yaw), sy = sinf(-yaw);
        float dx0 = p0.x - e0, dy0 = p0.y - e1;
        float dx1 = p1.x - e0, dy1 = p1.y - e1;
        p0.x = dx0 * cy - dy0 * sy;  p0.y = dx0 * sy + dy0 * cy;
        p1.x = dx1 * cy - dy1 * sy;  p1.y = dx1 * sy + dy1 * cy;
    }

    float ax = truncf((p0.x - (-20.0f)) / 0.1f);
    float ay = truncf((p0.y - (-10.0f)) / 0.1f);
    float bx = truncf((p1.x - (-20.0f)) / 0.1f);
    float by = truncf((p1.y - (-10.0f)) / 0.1f);

    bool va = (ax >= 0.0f) && (ax < (float)WW) && (ay >= 0.0f) && (ay < (float)HH);
    bool vb = (bx >= 0.0f) && (bx < (float)WW) && (by >= 0.0f) && (by < (float)HH);
    if (!(va || vb)) return;

    ax = fminf(fmaxf(ax, 0.0f), (float)(WW - 1));
    ay = fminf(fmaxf(ay, 0.0f), (float)(HH - 1));
    bx = fminf(fmaxf(bx, 0.0f), (float)(WW - 1));
    by = fminf(fmaxf(by, 0.0f), (float)(HH - 1));

    float dmax = fmaxf(fabsf(bx - ax), fabsf(by - ay));  // integer-valued
    float fk = (float)k;
    if (fk > dmax) return;                               // samples past dmax duplicate endpoint b
    float t = fminf(fk, dmax) / fmaxf(dmax, 1.0f);
    float fx = ax + t * (bx - ax);
    float fy = ay + t * (by - ay);
    int ix = (int)rintf(fx);                             // RNE, matches jnp.round
    int iy = (int)rintf(fy);
    #pragma unroll
    for (int dy = -1; dy <= 1; ++dy) {
        #pragma unroll
        for (int dx = -1; dx <= 1; ++dx) {
            int xx = ix + dx, yy = iy + dy;
            if (xx >= 0 && xx < WW && yy >= 0 && yy < HH)
                out_ch[yy * WW + xx] = 1.0f;             // max with 1.0; races benign
        }
    }
}

extern "C" void kernel_launch(void* const* d_in, const int* in_sizes, int n_in,
                              void* d_out, int out_size, void* d_ws, size_t ws_size,
                              hipStream_t stream) {
    (void)d_ws; (void)ws_size; (void)n_in; (void)out_size;
    const float4* lidar = (const float4*)d_in[0];
    const float2* traj  = (const float2*)d_in[1];
    const float2* osm   = (const float2*)d_in[2];
    const float*  ego   = (const float*)d_in[3];
    float* out = (float*)d_out;

    int npts  = in_sizes[0] / 4;
    int ntraj = in_sizes[1] / 2;
    int nosm  = in_sizes[2] / 2;

    int ntot = 5 * HWSZ;
    zero_kernel<<<(ntot + 255) / 256, 256, 0, stream>>>(out, ntot);

    lidar_scatter_kernel<<<(npts + 255) / 256, 256, 0, stream>>>(lidar, out, npts);
    finalize_kernel<<<(HWSZ + 255) / 256, 256, 0, stream>>>(out);

    if (ntraj > 1)
        polyline_kernel<<<ntraj - 1, 512, 0, stream>>>(traj, ntraj, out + 3 * HWSZ, ego, 0);
    if (nosm > 1)
        polyline_kernel<<<nosm - 1, 512, 0, stream>>>(osm, nosm, out + 4 * HWSZ, ego, 1);
}